// DGCNN_42228118454535
// MI455X (gfx1250) — compile-verified
//
#include <hip/hip_runtime.h>
#include <hip/hip_bf16.h>
#include <math.h>

#define NN 100000
#define NE 1600000
#define NG 256
#define HID 32
#define POOLW 192

typedef __attribute__((ext_vector_type(2))) float v2f;
typedef __attribute__((ext_vector_type(8))) float v8f;

// ---------------- utility ----------------
__global__ void k_zero(float* __restrict__ p, int n) {
    int i = blockIdx.x * blockDim.x + threadIdx.x;
    int stride = gridDim.x * blockDim.x;
    for (; i < n; i += stride) p[i] = 0.0f;
}

// degree count over edge destinations (self-loops added analytically later)
__global__ void k_deg(const int* __restrict__ dst, float* __restrict__ deg) {
    int e = blockIdx.x * blockDim.x + threadIdx.x;
    int stride = gridDim.x * blockDim.x;
    for (; e < NE; e += stride) atomicAdd(&deg[dst[e]], 1.0f);
}

// deg -> deg^-1/2 with self-loop (+1); deg+1 >= 1 always
__global__ void k_dinv(float* __restrict__ deg) {
    int i = blockIdx.x * blockDim.x + threadIdx.x;
    if (i < NN) deg[i] = rsqrtf(deg[i] + 1.0f);
}

// per-edge symmetric norm
__global__ void k_norm(const int* __restrict__ src, const int* __restrict__ dst,
                       const float* __restrict__ dinv, float* __restrict__ norm) {
    int e = blockIdx.x * blockDim.x + threadIdx.x;
    int stride = gridDim.x * blockDim.x;
    for (; e < NE; e += stride) norm[e] = dinv[src[e]] * dinv[dst[e]];
}

// ---------------- layer 1 feature transform (in_dim = 1) ----------------
__global__ void k_xw1(const float* __restrict__ x, const float* __restrict__ W,
                      float* __restrict__ t) {
    int idx = blockIdx.x * blockDim.x + threadIdx.x;   // N*32 threads
    if (idx >= NN * HID) return;
    int i = idx >> 5, f = idx & 31;
    t[idx] = x[i] * W[f];
}

// ---------------- layers 2..6 feature transform: t = h @ W  (N x 32 @ 32 x 32) --
// One wave computes a 16-row x 32-col slab via V_WMMA_F32_16X16X4_F32.
// Fragment layouts per CDNA5 ISA 7.12.2 (wave32):
//   A 16x4 : lane m(0..15)/half h : v0 = A[m][k0+2h], v1 = A[m][k0+2h+1]
//   B 4x16 : lane n(0..15)/half h : v0 = B[k0+2h][n], v1 = B[k0+2h+1][n]
//   C 16x16: lane n/half h        : v[i] = C[i + 8h][n]
__global__ void k_gemm32_wmma(const float* __restrict__ h, const float* __restrict__ W,
                              float* __restrict__ t) {
    const int NBLK = NN / 16;                                // 6250, exact
    int wid = (blockIdx.x * blockDim.x + threadIdx.x) >> 5;  // wave-uniform
    if (wid >= NBLK) return;                                 // whole wave exits
    int lane = threadIdx.x & 31;
    int half = lane >> 4;
    int m    = lane & 15;   // A row within tile
    int n    = lane & 15;   // B/C column within tile
    int r0   = wid * 16;

    v8f acc0 = {};  // columns 0..15
    v8f acc1 = {};  // columns 16..31
#pragma unroll
    for (int kk = 0; kk < 8; ++kk) {
        int k0 = kk * 4;
        v2f a;
        const float* ap = h + (r0 + m) * HID + k0 + 2 * half;
        a.x = ap[0];
        a.y = ap[1];
        int krow = k0 + 2 * half;
        v2f b0, b1;
        b0.x = W[krow * HID + n];
        b0.y = W[(krow + 1) * HID + n];
        b1.x = W[krow * HID + 16 + n];
        b1.y = W[(krow + 1) * HID + 16 + n];
        acc0 = __builtin_amdgcn_wmma_f32_16x16x4_f32(false, a, false, b0,
                                                     (short)0, acc0, false, false);
        acc1 = __builtin_amdgcn_wmma_f32_16x16x4_f32(false, a, false, b1,
                                                     (short)0, acc1, false, false);
    }
#pragma unroll
    for (int i = 0; i < 8; ++i) {
        int row = r0 + i + 8 * half;
        t[row * HID + n]      = acc0[i];
        t[row * HID + 16 + n] = acc1[i];
    }
}

// ---------------- out = bias + selfloop (dinv_i^2 * t_i) ----------------
__global__ void k_init_out(const float* __restrict__ t, const float* __restrict__ dinv,
                           const float* __restrict__ bias, float* __restrict__ out) {
    int idx = blockIdx.x * blockDim.x + threadIdx.x;   // N*32
    if (idx >= NN * HID) return;
    int i = idx >> 5, f = idx & 31;
    float di = dinv[i];
    out[idx] = bias[f] + di * di * t[idx];
}

// ---------------- edge scatter: out[dst] += norm[e] * t[src] ----------------
// One wave32 per edge; lane = feature. src/dst/norm are wave-broadcast loads,
// gather and atomic-add are 128B coalesced.
__global__ void k_scatter(const int* __restrict__ src, const int* __restrict__ dst,
                          const float* __restrict__ norm, const float* __restrict__ t,
                          float* __restrict__ out) {
    int idx = blockIdx.x * blockDim.x + threadIdx.x;   // NE*32, exact multiple of 256
    int e = idx >> 5, f = idx & 31;
    int s = src[e], d = dst[e];
    float v = norm[e] * t[s * HID + f];
    atomicAdd(&out[d * HID + f], v);
}

// ---------------- relu in place + graph pooling ----------------
__global__ void k_relu_pool(float* __restrict__ h, const int* __restrict__ batch,
                            float* __restrict__ pooled, int layer_off) {
    int idx = blockIdx.x * blockDim.x + threadIdx.x;   // N*32
    if (idx >= NN * HID) return;
    int i = idx >> 5, f = idx & 31;
    float v = h[idx];
    v = v > 0.0f ? v : 0.0f;
    h[idx] = v;
    atomicAdd(&pooled[batch[i] * POOLW + layer_off + f], v);
}

// ---------------- fc1: z1 = relu(pooled @ fc1_w + fc1_b) ----------------
__global__ void k_fc1(const float* __restrict__ pooled, const float* __restrict__ w,
                      const float* __restrict__ bias, float* __restrict__ z1) {
    int idx = blockIdx.x * blockDim.x + threadIdx.x;   // 256*128
    if (idx >= NG * 128) return;
    int g = idx >> 7, o = idx & 127;
    float acc = bias[o];
    const float* pg = pooled + g * POOLW;
#pragma unroll 4
    for (int k = 0; k < POOLW; ++k) acc += pg[k] * w[k * 128 + o];
    z1[idx] = acc > 0.0f ? acc : 0.0f;
}

// ---------------- fc2 + softmax ----------------
__global__ void k_fc2_softmax(const float* __restrict__ z1, const float* __restrict__ w,
                              const float* __restrict__ bias, float* __restrict__ out) {
    int g = blockIdx.x * blockDim.x + threadIdx.x;
    if (g >= NG) return;
    float a0 = bias[0], a1 = bias[1];
    const float* zg = z1 + g * 128;
#pragma unroll 4
    for (int k = 0; k < 128; ++k) {
        float z = zg[k];
        a0 += z * w[k * 2 + 0];
        a1 += z * w[k * 2 + 1];
    }
    float mx = fmaxf(a0, a1);
    float e0 = __expf(a0 - mx), e1 = __expf(a1 - mx);
    float inv = 1.0f / (e0 + e1);
    out[g * 2 + 0] = e0 * inv;
    out[g * 2 + 1] = e1 * inv;
}

extern "C" void kernel_launch(void* const* d_in, const int* in_sizes, int n_in,
                              void* d_out, int out_size, void* d_ws, size_t ws_size,
                              hipStream_t stream) {
    const float* x     = (const float*)d_in[0];
    const int*   ei    = (const int*)d_in[1];
    const int*   srcI  = ei;            // edge_index[0, :]
    const int*   dstI  = ei + NE;       // edge_index[1, :]
    const int*   batch = (const int*)d_in[2];
    const float* W[6];
    const float* B[6];
    for (int i = 0; i < 6; ++i) {
        W[i] = (const float*)d_in[3 + 2 * i];
        B[i] = (const float*)d_in[4 + 2 * i];
    }
    const float* fc1_w = (const float*)d_in[15];
    const float* fc1_b = (const float*)d_in[16];
    const float* fc2_w = (const float*)d_in[17];
    const float* fc2_b = (const float*)d_in[18];

    // workspace layout (floats)
    float* ws     = (float*)d_ws;
    float* dinv   = ws;                         // N
    float* norm   = dinv + NN;                  // E
    float* t      = norm + NE;                  // N*32
    float* ha     = t + (size_t)NN * HID;       // N*32
    float* hb     = ha + (size_t)NN * HID;      // N*32
    float* pooled = hb + (size_t)NN * HID;      // 256*192
    float* z1     = pooled + NG * POOLW;        // 256*128

    const int TB = 256;
    const int nNodeF = NN * HID;                       // 3.2M
    const int gNodeF = (nNodeF + TB - 1) / TB;         // 12500
    const int gN     = (NN + TB - 1) / TB;
    const int gE     = (NE + TB - 1) / TB;
    const int gEF    = (NE * HID) / TB;                // 200000, exact
    const int gWMMA  = ((NN / 16) * 32 + TB - 1) / TB; // waves=6250 -> 782 blocks

    // norm precompute
    k_zero<<<gN, TB, 0, stream>>>(dinv, NN);
    k_zero<<<(NG * POOLW + TB - 1) / TB, TB, 0, stream>>>(pooled, NG * POOLW);
    k_deg<<<gE, TB, 0, stream>>>(dstI, dinv);
    k_dinv<<<gN, TB, 0, stream>>>(dinv);
    k_norm<<<gE, TB, 0, stream>>>(srcI, dstI, dinv, norm);

    float* cur = nullptr;
    float* nxt = ha;
    for (int l = 0; l < 6; ++l) {
        if (l == 0)
            k_xw1<<<gNodeF, TB, 0, stream>>>(x, W[0], t);
        else
            k_gemm32_wmma<<<gWMMA, TB, 0, stream>>>(cur, W[l], t);
        k_init_out<<<gNodeF, TB, 0, stream>>>(t, dinv, B[l], nxt);
        k_scatter<<<gEF, TB, 0, stream>>>(srcI, dstI, norm, t, nxt);
        k_relu_pool<<<gNodeF, TB, 0, stream>>>(nxt, batch, pooled, l * HID);
        cur = nxt;
        nxt = (cur == ha) ? hb : ha;
    }

    k_fc1<<<(NG * 128 + TB - 1) / TB, TB, 0, stream>>>(pooled, fc1_w, fc1_b, z1);
    k_fc2_softmax<<<1, NG, 0, stream>>>(z1, fc2_w, fc2_b, (float*)d_out);
}